// EGATConv_8607114461242
// MI455X (gfx1250) — compile-verified
//
#include <hip/hip_runtime.h>
#include <hip/hip_bf16.h>
#include <cstdint>

#define NODES 50000
#define EDGES 400000
#define INCH  256
#define EDIM  64
#define HEADS 4
#define OUTC  64
#define OC    256   // HEADS*OUTC
#define NEG_SLOPE 0.2f

typedef __attribute__((ext_vector_type(2))) float v2f;
typedef __attribute__((ext_vector_type(8))) float v8f;

// ---- order-preserving float <-> uint key for atomic segment-max ----
__device__ __forceinline__ unsigned fkey(float f) {
    unsigned u = __float_as_uint(f);
    return (u & 0x80000000u) ? ~u : (u | 0x80000000u);
}
__device__ __forceinline__ float fdec(unsigned k) {
    unsigned u = (k & 0x80000000u) ? (k ^ 0x80000000u) : ~k;
    return __uint_as_float(u);
}

// ============================================================
// Kernel 1: xh = x @ W_lin^T via fp32 WMMA 16x16x4.
// One wave computes one 16x16 output tile. 3125*16 tiles, 8 waves/block.
// A[m,k] = x[m0+m, k];  B[k,n] = W_lin[n0+n, k]  (both K-major fp32 rows).
// ============================================================
__global__ __launch_bounds__(256) void gemm_xh_kernel(
        const float* __restrict__ x, const float* __restrict__ Wlin,
        float* __restrict__ xh) {
    const int wave = (blockIdx.x * blockDim.x + threadIdx.x) >> 5;
    const int lane = threadIdx.x & 31;
    const int tileM = wave >> 4;        // / (OC/16)
    const int tileN = wave & 15;
    const int m0 = tileM * 16, n0 = tileN * 16;
    const int lid  = lane & 15;
    const int half = lane >> 4;

    const float* __restrict__ arow = x    + (size_t)(m0 + lid) * INCH;
    const float* __restrict__ brow = Wlin + (size_t)(n0 + lid) * INCH;

    v8f c = {};
#pragma unroll 8
    for (int k = 0; k < INCH; k += 4) {
        const int kb = k + half * 2;                 // even -> 8B aligned
        v2f a = *(const v2f*)(arow + kb);            // A frag: VGPR0=K=kb, VGPR1=K=kb+1
        v2f b = *(const v2f*)(brow + kb);            // B frag: same K striping, N=lid
        c = __builtin_amdgcn_wmma_f32_16x16x4_f32(
                false, a, false, b, (short)0, c, false, false);
    }

    float* __restrict__ op = xh + (size_t)m0 * OC + n0 + lid;
#pragma unroll
    for (int v = 0; v < 8; ++v) {
        op[(size_t)(v + half * 8) * OC] = c[v];      // D: VGPR v -> rows M=v, M=v+8
    }
}

// ============================================================
// Kernel 2: M_edge[d*4+h] = sum_c a_e[h,c] * W_edge[h*64+c, d]
// (folds the E x 256 edge GEMM into a 64x4 matrix). One block.
// ============================================================
__global__ void prep_medge_kernel(const float* __restrict__ We,
                                  const float* __restrict__ att,
                                  float* __restrict__ Me) {
    const int t = threadIdx.x;       // 256 threads
    const int d = t >> 2, h = t & 3;
    float acc = 0.f;
#pragma unroll 8
    for (int c = 0; c < OUTC; ++c)
        acc += att[h * 192 + 128 + c] * We[(size_t)(h * 64 + c) * EDIM + d];
    Me[d * 4 + h] = acc;
}

// ============================================================
// Kernel 3: per-node attention scores s_i[n,h] = xh[n,h,:]·a_i[h],
//           s_j likewise. One wave (32 lanes) per node, shuffle-reduce.
// ============================================================
__global__ __launch_bounds__(256) void node_scores_kernel(
        const float* __restrict__ xh, const float* __restrict__ att,
        float* __restrict__ s_i, float* __restrict__ s_j) {
    const int node = (blockIdx.x * blockDim.x + threadIdx.x) >> 5;
    const int lane = threadIdx.x & 31;
    const float* __restrict__ xr = xh + (size_t)node * OC;

    float ai[HEADS] = {0, 0, 0, 0}, aj[HEADS] = {0, 0, 0, 0};
#pragma unroll
    for (int it = 0; it < 8; ++it) {
        const int h  = it >> 1;               // 32 consecutive channels share a head
        const int ch = lane + it * 32;
        const int cc = ch & 63;
        const float v = xr[ch];
        ai[h] += v * att[h * 192 + cc];
        aj[h] += v * att[h * 192 + 64 + cc];
    }
#pragma unroll
    for (int h = 0; h < HEADS; ++h) {
#pragma unroll
        for (int off = 16; off > 0; off >>= 1) {
            ai[h] += __shfl_xor(ai[h], off, 32);
            aj[h] += __shfl_xor(aj[h], off, 32);
        }
        if (lane == 0) {
            s_i[node * HEADS + h] = ai[h];
            s_j[node * HEADS + h] = aj[h];
        }
    }
}

// ============================================================
// Kernel 4: per-edge logits: ed = edge_attr[e]·M_edge, alpha = leaky_relu(
//           s_i[row] + s_j[col] + ed), then atomic segment-max on row.
// ============================================================
__global__ __launch_bounds__(256) void edge_scores_kernel(
        const float* __restrict__ ea, const long long* __restrict__ ei,
        const float* __restrict__ s_i, const float* __restrict__ s_j,
        const float* __restrict__ Me, float* __restrict__ alpha,
        unsigned* __restrict__ nmax) {
    __shared__ float sMe[EDIM * HEADS];
    sMe[threadIdx.x] = Me[threadIdx.x];
    __syncthreads();

    const int e = blockIdx.x * blockDim.x + threadIdx.x;
    if (e >= EDGES) return;
    const int r  = (int)ei[e];
    const int cl = (int)ei[EDGES + e];

    float acc[HEADS] = {0, 0, 0, 0};
    const float4* __restrict__ row = (const float4*)(ea + (size_t)e * EDIM);
#pragma unroll 4
    for (int d4 = 0; d4 < EDIM / 4; ++d4) {
        const float4 v = row[d4];
        const int d = d4 * 4;
#pragma unroll
        for (int h = 0; h < HEADS; ++h)
            acc[h] += v.x * sMe[(d + 0) * 4 + h] + v.y * sMe[(d + 1) * 4 + h] +
                      v.z * sMe[(d + 2) * 4 + h] + v.w * sMe[(d + 3) * 4 + h];
    }
#pragma unroll
    for (int h = 0; h < HEADS; ++h) {
        float a = s_i[r * HEADS + h] + s_j[cl * HEADS + h] + acc[h];
        a = (a > 0.f) ? a : NEG_SLOPE * a;          // leaky_relu
        alpha[(size_t)e * HEADS + h] = a;
        atomicMax(&nmax[r * HEADS + h], fkey(a));
    }
}

// ============================================================
// Kernel 5: alpha = exp(alpha - max[row]); denom[row] += alpha (atomic)
// ============================================================
__global__ __launch_bounds__(256) void exp_denom_kernel(
        const long long* __restrict__ ei, const unsigned* __restrict__ nmax,
        float* __restrict__ alpha, float* __restrict__ denom) {
    const int idx = blockIdx.x * blockDim.x + threadIdx.x;  // E*HEADS threads
    const int e = idx >> 2, h = idx & 3;
    const int r = (int)ei[e];
    const float m = fdec(nmax[r * HEADS + h]);
    const float w = __expf(alpha[idx] - m);
    alpha[idx] = w;
    atomicAdd(&denom[r * HEADS + h], w);
}

// ============================================================
// Kernel 6: scatter messages: acc[row, c] += xh[col, c] * alpha/(denom+eps).
// One block per edge (256 threads = 256 channels); xh & acc are L2-resident.
// ============================================================
__global__ __launch_bounds__(256) void scatter_kernel(
        const long long* __restrict__ ei, const float* __restrict__ alpha,
        const float* __restrict__ denom, const float* __restrict__ xh,
        float* __restrict__ acc) {
    const int e = blockIdx.x;
    const int c = threadIdx.x;
    const int h = c >> 6;
    const int r  = (int)ei[e];
    const int cl = (int)ei[EDGES + e];
    const float coef = alpha[(size_t)e * HEADS + h] /
                       (denom[r * HEADS + h] + 1e-16f);
    atomicAdd(&acc[(size_t)r * OC + c], xh[(size_t)cl * OC + c] * coef);
}

// ============================================================
// Kernel 7: mean over heads + bias
// ============================================================
__global__ __launch_bounds__(256) void finalize_kernel(
        const float* __restrict__ acc, const float* __restrict__ bias,
        float* __restrict__ out) {
    const int idx = blockIdx.x * blockDim.x + threadIdx.x;  // N*OUTC
    const int n = idx >> 6, c = idx & 63;
    const float* __restrict__ a = acc + (size_t)n * OC;
    out[idx] = 0.25f * (a[c] + a[64 + c] + a[128 + c] + a[192 + c]) + bias[c];
}

extern "C" void kernel_launch(void* const* d_in, const int* in_sizes, int n_in,
                              void* d_out, int out_size, void* d_ws, size_t ws_size,
                              hipStream_t stream) {
    const float*     x    = (const float*)d_in[0];
    const long long* ei   = (const long long*)d_in[1];   // int64 [2, E]
    const float*     ea   = (const float*)d_in[2];
    const float*     Wlin = (const float*)d_in[3];
    const float*     We   = (const float*)d_in[4];
    const float*     att  = (const float*)d_in[5];
    const float*     bias = (const float*)d_in[6];
    float*           out  = (float*)d_out;

    char* ws = (char*)d_ws;
    float*    xh    = (float*)ws;    ws += (size_t)NODES * OC * 4;
    float*    acc   = (float*)ws;    ws += (size_t)NODES * OC * 4;
    float*    alpha = (float*)ws;    ws += (size_t)EDGES * HEADS * 4;
    float*    s_i   = (float*)ws;    ws += (size_t)NODES * HEADS * 4;
    float*    s_j   = (float*)ws;    ws += (size_t)NODES * HEADS * 4;
    unsigned* nmax  = (unsigned*)ws; ws += (size_t)NODES * HEADS * 4;
    float*    denom = (float*)ws;    ws += (size_t)NODES * HEADS * 4;
    float*    Me    = (float*)ws;    ws += 1024;

    hipMemsetAsync(acc,   0, (size_t)NODES * OC * 4, stream);
    hipMemsetAsync(nmax,  0, (size_t)NODES * HEADS * 4, stream);
    hipMemsetAsync(denom, 0, (size_t)NODES * HEADS * 4, stream);

    // 3125*16 tiles / 8 waves per block = 6250 blocks (exact, no tails)
    gemm_xh_kernel<<<(NODES / 16) * (OC / 16) / 8, 256, 0, stream>>>(x, Wlin, xh);
    prep_medge_kernel<<<1, 256, 0, stream>>>(We, att, Me);
    node_scores_kernel<<<NODES / 8, 256, 0, stream>>>(xh, att, s_i, s_j);
    edge_scores_kernel<<<(EDGES + 255) / 256, 256, 0, stream>>>(ea, ei, s_i, s_j,
                                                                Me, alpha, nmax);
    exp_denom_kernel<<<EDGES * HEADS / 256, 256, 0, stream>>>(ei, nmax, alpha, denom);
    scatter_kernel<<<EDGES, 256, 0, stream>>>(ei, alpha, denom, xh, acc);
    finalize_kernel<<<NODES * OUTC / 256, 256, 0, stream>>>(acc, bias, out);
}